// Posmap2Normal_53858889892198
// MI455X (gfx1250) — compile-verified
//
#include <hip/hip_runtime.h>
#include <hip/hip_bf16.h>

// Posmap -> surface normal stencil (b=3, 3ch, 1536x1536). Memory-bound:
// ~177 MB HBM traffic => ~7.6us floor at 23.3 TB/s. Per 256-thread block,
// a 32x8 output tile is computed from a 34x10x3 posmap halo tile in LDS.
//
// CDNA5 data paths:
//  - Interior blocks (~95%): one TDM tensor_load_to_lds (3D tile descriptor
//    per ISA 8.3-8.6) issued by wave 0, waited with s_wait_tensorcnt.
//  - Boundary blocks: per-lane global_load_async_to_lds_b32 with edge-clamped
//    addresses (out-of-image posmap values are always multiplied by a zero
//    mask-validity factor, so only the mask halo needs exact zeros).
//  - Mask halo staged branchlessly (clamped load * in-bounds flag).

#define HH 1536
#define WW 1536
#define TILE_W 32
#define TILE_H 8
#define LW (TILE_W + 2)   // 34
#define LH (TILE_H + 2)   // 10
#define LN (LW * LH)      // 340 halo elements
#define NTHREADS 256

typedef __attribute__((ext_vector_type(4))) unsigned int uint32x4;
typedef __attribute__((ext_vector_type(8))) int          int32x8;
typedef __attribute__((ext_vector_type(4))) int          int32x4;

__global__ __launch_bounds__(NTHREADS) void posmap2normal_kernel(
    const float* __restrict__ posmap,        // [B,3,H,W] f32
    const unsigned char* __restrict__ mask,  // [B,H,W] bool (1 byte)
    float* __restrict__ out)                 // [B,3,H,W] f32
{
    __shared__ float sP[3][LH][LW];  // 4080 B posmap halo tile (TDM fills contiguously X->Y->Z)
    __shared__ float sM[LH][LW];     // 1360 B mask halo tile

    const int tid = threadIdx.x;
    const int x0 = blockIdx.x * TILE_W;
    const int y0 = blockIdx.y * TILE_H;
    const int bb = blockIdx.z;

    const size_t plane = (size_t)HH * WW;
    const float* pBase = posmap + (size_t)bb * 3 * plane;
    const unsigned char* mBase = mask + (size_t)bb * plane;

    const bool interior = (x0 > 0) & (x0 + TILE_W < WW) & (y0 > 0) & (y0 + TILE_H < HH);

    // ---- Stage posmap halo tile ----
    if (interior) {
        if (tid < 32) {  // wave 0 issues one TDM 3D-tile load: 34 x 10 x 3 f32
            const unsigned long long gaddr =
                (unsigned long long)(uintptr_t)(pBase + (size_t)(y0 - 1) * WW + (x0 - 1));
            const unsigned ldsBase = (unsigned)(uintptr_t)&sP[0][0][0];

            uint32x4 g0;
            g0[0] = 1u;                                   // count=1, user descriptor
            g0[1] = ldsBase;                              // lds_addr (bytes)
            g0[2] = (unsigned)(gaddr & 0xFFFFFFFFull);    // global_addr[31:0]
            g0[3] = (unsigned)((gaddr >> 32) & 0x1FFFFFFull) | (2u << 30);  // addr[56:32] | type=2

            int32x8 g1;
            g1[0] = 0x00020000;                 // wg_mask=0, data_size=2 (4B), no pad/iterate
            g1[1] = (int)((unsigned)WW << 16);  // tensor_dim0[15:0] (dim0 = W)
            g1[2] = (int)((unsigned)HH << 16);  // tensor_dim0[31:16]=0 | tensor_dim1[15:0] (H)
            g1[3] = (int)(34u << 16);           // tensor_dim1[31:16]=0 | tile_dim0=34
            g1[4] = (int)((3u << 16) | 10u);    // tile_dim1=10 | tile_dim2=3
            g1[5] = WW;                         // tensor_dim0_stride[31:0] = W
            g1[6] = (int)(((unsigned)plane & 0xFFFFu) << 16);   // str0[47:32]=0 | str1[15:0]
            g1[7] = (int)((unsigned)plane >> 16);               // tensor_dim1_stride[47:16]

            int32x4 g2;
            g2[0] = 3;                          // tensor_dim2 = 3 channels
            g2[1] = 1;                          // tensor_dim3 (unused, nonzero-safe)
            g2[2] = (int)(3u * (unsigned)plane);// tensor_dim2_stride[31:0] = 3*H*W
            g2[3] = 0;                          // str2[47:32]=0 | tile_dim3=0

            int32x4 g3;
            g3[0] = 0;                          // tensor_dim3_stride[31:0]
            g3[1] = (int)(1u << 16);            // str3[47:32]=0 | tensor_dim4=1
            g3[2] = 0;                          // tensor_dim4[31:16]=0 | tile_dim4=0
            g3[3] = 0;

#if defined(__clang_major__) && __clang_major__ >= 23
            // amdgpu-toolchain (clang-23 / therock-10.0): 6-arg builtin; the
            // extra int32x8 maps to the unused VADDR4 operand slot -> zeros.
            int32x8 g4 = {0, 0, 0, 0, 0, 0, 0, 0};
            __builtin_amdgcn_tensor_load_to_lds(g0, g1, g2, g3, g4, 0);
#else
            // ROCm 7.2 (clang-22): 5-arg builtin.
            __builtin_amdgcn_tensor_load_to_lds(g0, g1, g2, g3, 0);
#endif
        }
    } else {
        // Boundary: per-lane async global->LDS copies, edge-clamped addresses.
        #pragma unroll
        for (int e0 = tid; e0 < 2 * NTHREADS; e0 += NTHREADS) {
            int e = (e0 < LN) ? e0 : (LN - 1);   // keep EXEC full; dup writes benign
            int ly = e / LW;
            int lx = e - ly * LW;
            int gy = y0 - 1 + ly;
            int gx = x0 - 1 + lx;
            int cy = gy < 0 ? 0 : (gy > HH - 1 ? HH - 1 : gy);
            int cx = gx < 0 ? 0 : (gx > WW - 1 ? WW - 1 : gx);
            size_t eidx = (size_t)cy * WW + cx;
            #pragma unroll
            for (int c = 0; c < 3; ++c) {
                unsigned long long ga =
                    (unsigned long long)(uintptr_t)(pBase + (size_t)c * plane + eidx);
                unsigned lo = (unsigned)(uintptr_t)&sP[c][ly][lx];  // low32 == DS offset
                asm volatile("global_load_async_to_lds_b32 %0, %1, off"
                             :: "v"(lo), "v"(ga) : "memory");
            }
        }
    }

    // ---- Stage mask halo (branchless; exact zeros outside image) ----
    // Overlaps with the in-flight TDM / async copies above.
    #pragma unroll
    for (int e0 = tid; e0 < 2 * NTHREADS; e0 += NTHREADS) {
        int e = (e0 < LN) ? e0 : (LN - 1);
        int ly = e / LW;
        int lx = e - ly * LW;
        int gy = y0 - 1 + ly;
        int gx = x0 - 1 + lx;
        int cy = gy < 0 ? 0 : (gy > HH - 1 ? HH - 1 : gy);
        int cx = gx < 0 ? 0 : (gx > WW - 1 ? WW - 1 : gx);
        float inb = ((gy >= 0) & (gy < HH) & (gx >= 0) & (gx < WW)) ? 1.0f : 0.0f;
        sM[ly][lx] = inb * (float)mBase[(size_t)cy * WW + cx];
    }

    if (interior) {
        __builtin_amdgcn_s_wait_tensorcnt(0);   // waves 1-7 have TENSORcnt==0: no-op
    } else {
        asm volatile("s_wait_asynccnt 0" ::: "memory");
    }
    __syncthreads();

    // ---- Compute one output pixel per thread ----
    const int tx = tid & (TILE_W - 1);
    const int ty = tid >> 5;
    const int lx = tx + 1;
    const int ly = ty + 1;

    float dR[3], dU[3], dL[3], dD[3];
    #pragma unroll
    for (int c = 0; c < 3; ++c) {
        float pc = sP[c][ly][lx];
        dR[c] = sP[c][ly][lx + 1] - pc;   // tap0: right
        dU[c] = sP[c][ly - 1][lx] - pc;   // tap1: up
        dL[c] = sP[c][ly][lx - 1] - pc;   // tap2: left
        dD[c] = sP[c][ly + 1][lx] - pc;   // tap3: down
    }

    float m  = sM[ly][lx];
    float mu = sM[ly - 1][lx];
    float md = sM[ly + 1][lx];
    float ml = sM[ly][lx - 1];
    float mr = sM[ly][lx + 1];
    float v0 = m * mu * mr;   // taps (0,1)
    float v1 = m * mu * ml;   // taps (1,2)
    float v2 = m * ml * md;   // taps (2,3)
    float v3 = m * mr * md;   // taps (3,0)

    // cross(a,b) = (a1*b2-a2*b1, a2*b0-a0*b2, a0*b1-a1*b0)
    float n0 = v0 * (dR[1]*dU[2] - dR[2]*dU[1])
             + v1 * (dU[1]*dL[2] - dU[2]*dL[1])
             + v2 * (dL[1]*dD[2] - dL[2]*dD[1])
             + v3 * (dD[1]*dR[2] - dD[2]*dR[1]);
    float n1 = v0 * (dR[2]*dU[0] - dR[0]*dU[2])
             + v1 * (dU[2]*dL[0] - dU[0]*dL[2])
             + v2 * (dL[2]*dD[0] - dL[0]*dD[2])
             + v3 * (dD[2]*dR[0] - dD[0]*dR[2]);
    float n2 = v0 * (dR[0]*dU[1] - dR[1]*dU[0])
             + v1 * (dU[0]*dL[1] - dU[1]*dL[0])
             + v2 * (dL[0]*dD[1] - dL[1]*dD[0])
             + v3 * (dD[0]*dR[1] - dD[1]*dR[0]);

    float nn = n0 * n0 + n1 * n1 + n2 * n2;
    float inv = 1.0f / fmaxf(sqrtf(nn), 1e-12f);

    size_t o = (size_t)bb * 3 * plane + (size_t)(y0 + ty) * WW + (size_t)(x0 + tx);
    out[o]             = n0 * inv;
    out[o + plane]     = n1 * inv;
    out[o + 2 * plane] = n2 * inv;
}

extern "C" void kernel_launch(void* const* d_in, const int* in_sizes, int n_in,
                              void* d_out, int out_size, void* d_ws, size_t ws_size,
                              hipStream_t stream) {
    (void)in_sizes; (void)n_in; (void)out_size; (void)d_ws; (void)ws_size;
    const float* posmap = (const float*)d_in[0];
    const unsigned char* maskp = (const unsigned char*)d_in[1];  // jnp.bool_: 1 byte/elem
    float* out = (float*)d_out;

    dim3 grid(WW / TILE_W, HH / TILE_H, 3);  // 48 x 192 x 3 = 27648 blocks
    posmap2normal_kernel<<<grid, NTHREADS, 0, stream>>>(posmap, maskp, out);
}